// LinearDecoder_70463233458487
// MI455X (gfx1250) — compile-verified
//
#include <hip/hip_runtime.h>

// NRI linear decoder, collapsed to dense recurrence:
//   R_b[r][s] = sampled_edges[b, e(s,r), 1]   (128x128, zero diag)
//   H_{t+1}   = H_t @ Ws^T + (R_b @ H_t) @ We^T,   H: 128x32, T=64 steps
// One workgroup per batch; R, H (double-buffered), M, Ws^T, We^T in LDS.
// Loop-invariant R / W^T fragments are hoisted into registers before the
// 64-step recurrence. f32 matrix math via V_WMMA_F32_16X16X4_F32 (wave32).

typedef __attribute__((ext_vector_type(2))) float v2f;
typedef __attribute__((ext_vector_type(4))) float v4f;
typedef __attribute__((ext_vector_type(8))) float v8f;

#define NVAR   128
#define FDIM   32
#define TSTEPS 64
#define NEDGE  16256            // 128*127
#define RSTR   132              // padded row stride for R (floats)
#define HSTR   36               // padded row stride for H/M/W (floats)

// LDS layout (in floats)
#define R_OFF   0               // 128*132 = 16896
#define H0_OFF  16896           // 128*36  = 4608
#define H1_OFF  21504           // 128*36  = 4608
#define M_OFF   26112           // 128*36  = 4608
#define WS_OFF  30720           // 32*36   = 1152
#define WE_OFF  31872           // 32*36   = 1152
#define LDS_FLOATS 33024        // 132,096 bytes

__global__ __launch_bounds__(256)
void nri_lindec_wmma_kernel(const float* __restrict__ inputs,
                            const float* __restrict__ rel,
                            const float* __restrict__ W_edge,
                            const float* __restrict__ W_self,
                            float* __restrict__ out) {
  extern __shared__ float lds[];
  const int b    = blockIdx.x;
  const int tid  = threadIdx.x;
  const int lane = tid & 31;
  const int wave = tid >> 5;          // 0..7
  const int half = lane >> 4;         // 0 or 1
  const int lm   = lane & 15;

  // ---------------- setup: stage R, W^T, H0 into LDS ----------------
  // R[r][s] = rel[b, s*127 + (r<s ? r : r-1), 1], zero diagonal
  for (int idx = tid; idx < NVAR * NVAR; idx += 256) {
    const int r = idx >> 7;
    const int s = idx & 127;
    float v = 0.0f;
    if (r != s) {
      const int e = s * 127 + (r < s ? r : r - 1);
      v = rel[((size_t)b * NEDGE + e) * 2 + 1];
    }
    lds[R_OFF + r * RSTR + s] = v;
  }
  // Transposed weights: WsT[f][g] = W_self[g][f]; WeT[f][g] = W_edge[1][g][f]
  {
    const int g = tid >> 5;           // 0..7 per pass
    const int f = tid & 31;
    #pragma unroll
    for (int gg = 0; gg < 32; gg += 8) {
      lds[WS_OFF + f * HSTR + (g + gg)] = W_self[(g + gg) * 32 + f];
      lds[WE_OFF + f * HSTR + (g + gg)] = W_edge[1024 + (g + gg) * 32 + f];
    }
  }
  // H0 = inputs[b, 0]  (128x32, coalesced float4)
  {
    const float* src = inputs + (size_t)b * TSTEPS * NVAR * FDIM;
    for (int i4 = tid; i4 < (NVAR * FDIM) / 4; i4 += 256) {
      const int i = i4 * 4;
      const int n = i >> 5;
      const int f = i & 31;
      *(v4f*)(lds + H0_OFF + n * HSTR + f) = *(const v4f*)(src + i);
    }
  }
  __syncthreads();

  const int rowA = wave * 16 + lm;    // this lane's A-fragment row

  // ---- hoist loop-invariant fragments into registers ----
  // R A-fragments for GEMM1: 32 k-steps x v2f  (constant across t)
  v2f rf[32];
  {
    const float* Ra = lds + R_OFF + rowA * RSTR + 2 * half;
    #pragma unroll
    for (int kk = 0; kk < 32; ++kk)
      rf[kk] = *(const v2f*)(Ra + 4 * kk);
  }
  // W^T B-fragments for GEMM2: 8 k-steps x 2 N-tiles x 2 matrices
  v2f ws0[8], ws1[8], we0[8], we1[8];
  {
    #pragma unroll
    for (int kk = 0; kk < 8; ++kk) {
      const float* Bs = lds + WS_OFF + (4 * kk + 2 * half) * HSTR;
      ws0[kk].x = Bs[lm];        ws0[kk].y = Bs[HSTR + lm];
      ws1[kk].x = Bs[16 + lm];   ws1[kk].y = Bs[HSTR + 16 + lm];
      const float* Be = lds + WE_OFF + (4 * kk + 2 * half) * HSTR;
      we0[kk].x = Be[lm];        we0[kk].y = Be[HSTR + lm];
      we1[kk].x = Be[16 + lm];   we1[kk].y = Be[HSTR + 16 + lm];
    }
  }

  int hc = H0_OFF;                    // current state buffer
  int hn = H1_OFF;                    // next state buffer
  float* outb = out + (size_t)b * TSTEPS * NVAR * FDIM;

  for (int t = 0; t < TSTEPS; ++t) {
    // ---------- Phase A: M = R @ H   (128x128 @ 128x32) ----------
    v8f m0 = {}; v8f m1 = {};
    {
      #pragma unroll
      for (int kk = 0; kk < 32; ++kk) {
        const float* Hk = lds + hc + (4 * kk + 2 * half) * HSTR;
        v2f b0, b1;
        b0.x = Hk[lm];        b0.y = Hk[HSTR + lm];      // B: 4x16, N tile 0
        b1.x = Hk[16 + lm];   b1.y = Hk[HSTR + 16 + lm]; // B: 4x16, N tile 1
        m0 = __builtin_amdgcn_wmma_f32_16x16x4_f32(false, rf[kk], false, b0,
                                                   (short)0, m0, false, false);
        m1 = __builtin_amdgcn_wmma_f32_16x16x4_f32(false, rf[kk], false, b1,
                                                   (short)0, m1, false, false);
      }
    }
    // spill M tiles to LDS (C/D layout: VGPR v -> row v + 8*half)
    {
      float* Mr = lds + M_OFF + (wave * 16 + 8 * half) * HSTR + lm;
      #pragma unroll
      for (int v = 0; v < 8; ++v) {
        Mr[v * HSTR]      = m0[v];
        Mr[v * HSTR + 16] = m1[v];
      }
    }
    __syncthreads();

    // ---------- Phase B: P = H @ WsT + M @ WeT  (K = 32 each) ----------
    v8f p0 = {}; v8f p1 = {};
    {
      const float* Ah = lds + hc    + rowA * HSTR + 2 * half;
      const float* Am = lds + M_OFF + rowA * HSTR + 2 * half;
      #pragma unroll
      for (int kk = 0; kk < 8; ++kk) {
        const v2f a = *(const v2f*)(Ah + 4 * kk);
        p0 = __builtin_amdgcn_wmma_f32_16x16x4_f32(false, a, false, ws0[kk],
                                                   (short)0, p0, false, false);
        p1 = __builtin_amdgcn_wmma_f32_16x16x4_f32(false, a, false, ws1[kk],
                                                   (short)0, p1, false, false);
      }
      #pragma unroll
      for (int kk = 0; kk < 8; ++kk) {
        const v2f a = *(const v2f*)(Am + 4 * kk);
        p0 = __builtin_amdgcn_wmma_f32_16x16x4_f32(false, a, false, we0[kk],
                                                   (short)0, p0, false, false);
        p1 = __builtin_amdgcn_wmma_f32_16x16x4_f32(false, a, false, we1[kk],
                                                   (short)0, p1, false, false);
      }
    }
    // store P -> H_next
    {
      float* Hr = lds + hn + (wave * 16 + 8 * half) * HSTR + lm;
      #pragma unroll
      for (int v = 0; v < 8; ++v) {
        Hr[v * HSTR]      = p0[v];
        Hr[v * HSTR + 16] = p1[v];
      }
    }
    __syncthreads();

    // coalesced writeback of the new state to out[b, t]
    {
      float* dst = outb + (size_t)t * NVAR * FDIM;
      for (int i4 = tid; i4 < (NVAR * FDIM) / 4; i4 += 256) {
        const int i = i4 * 4;
        const int n = i >> 5;
        const int f = i & 31;
        *(v4f*)(dst + i) = *(const v4f*)(lds + hn + n * HSTR + f);
      }
    }
    // swap state buffers
    const int tmp = hc; hc = hn; hn = tmp;
  }
}

extern "C" void kernel_launch(void* const* d_in, const int* in_sizes, int n_in,
                              void* d_out, int out_size, void* d_ws, size_t ws_size,
                              hipStream_t stream) {
  const float* inputs = (const float*)d_in[0];   // (32,64,128,32)
  const float* rel    = (const float*)d_in[1];   // (32,16256,2)
  const float* W_edge = (const float*)d_in[2];   // (2,32,32)
  const float* W_self = (const float*)d_in[3];   // (32,32)
  float* out = (float*)d_out;                    // (32,64,128,32)

  const size_t lds_bytes = LDS_FLOATS * sizeof(float);
  (void)hipFuncSetAttribute((const void*)nri_lindec_wmma_kernel,
                            hipFuncAttributeMaxDynamicSharedMemorySize,
                            (int)lds_bytes);

  nri_lindec_wmma_kernel<<<dim3(32), dim3(256), lds_bytes, stream>>>(
      inputs, rel, W_edge, W_self, out);
}